// Attn_13357348290853
// MI455X (gfx1250) — compile-verified
//
#include <hip/hip_runtime.h>
#include <math.h>

// Shapes from the reference
#define S_LEN 4096
#define BATCH 32
#define HDIM  512

typedef __attribute__((ext_vector_type(2))) float v2f;
typedef __attribute__((ext_vector_type(4))) float v4f;
typedef __attribute__((ext_vector_type(8))) float v8f;

// ---------------------------------------------------------------------------
// Kernel 1: v[b,h] = sum_g hidden[b,g] * W[g,h]   (32x512 = 2x32 WMMA tiles)
// Exact f32 GEMM via V_WMMA_F32_16X16X4_F32. One wave per 16x16 D tile.
// A-frag (16x4 f32, ISA 7.12.2): lane<16 holds {K=0,K=1}, lane>=16 holds {K=2,K=3},
// row M = lane%16. B-frag (4x16) mirrored: col N = lane%16, same K split.
// D-frag: VGPR r -> row r + 8*(lane>=16), col = lane%16.
// ---------------------------------------------------------------------------
__global__ __launch_bounds__(32)
void precompute_v_kernel(const float* __restrict__ hidden,   // [B,H]
                         const float* __restrict__ W,        // [H,H] (g-major)
                         float* __restrict__ v) {             // [B,H]
    const int tileN = blockIdx.x;          // 0..31  (H/16)
    const int tileM = blockIdx.y;          // 0..1   (B/16)
    const int lane  = threadIdx.x;         // 0..31
    const int half  = lane >> 4;           // 0 or 1
    const int l16   = lane & 15;

    const int m = tileM * 16 + l16;        // A row (batch index)
    const int n = tileN * 16 + l16;        // B col (h index)

    v8f acc = {};
    for (int k = 0; k < HDIM; k += 4) {
        const int kk = k + half * 2;
        v2f a, b;
        a.x = hidden[m * HDIM + kk];
        a.y = hidden[m * HDIM + kk + 1];
        b.x = W[(size_t)kk * HDIM + n];
        b.y = W[(size_t)(kk + 1) * HDIM + n];
        acc = __builtin_amdgcn_wmma_f32_16x16x4_f32(
            /*neg_a=*/false, a, /*neg_b=*/false, b,
            /*c_mod=*/(short)0, acc, /*reuse_a=*/false, /*reuse_b=*/false);
    }

#pragma unroll
    for (int r = 0; r < 8; ++r) {
        const int row = tileM * 16 + half * 8 + r;
        v[row * HDIM + n] = acc[r];
    }
}

// ---------------------------------------------------------------------------
// Kernel 2: c[b] = hidden[b,:] . bias   (one wave per b; 1 block of 32 waves)
// ---------------------------------------------------------------------------
__global__ __launch_bounds__(1024)
void precompute_c_kernel(const float* __restrict__ hidden,
                         const float* __restrict__ bias,
                         float* __restrict__ c) {
    const int b    = threadIdx.x >> 5;
    const int lane = threadIdx.x & 31;
    float s = 0.f;
#pragma unroll
    for (int i = lane; i < HDIM; i += 32)
        s += hidden[b * HDIM + i] * bias[i];
#pragma unroll
    for (int off = 16; off > 0; off >>= 1)
        s += __shfl_xor(s, off, 32);
    if (lane == 0) c[b] = s;
}

// ---------------------------------------------------------------------------
// Kernel 3 (the HBM-bound one): scores[b,s] = enc[s,b,:] . v[b,:] + c[b]
// One wave per (s,b): reads 2 KB contiguous; 8 consecutive b per block ->
// 16 KB contiguous per block. Non-temporal loads for the 256 MB single-use
// stream; v (64 KB) stays L2-hot. Stores scores transposed to [B,S] for the
// softmax pass.
// ---------------------------------------------------------------------------
__global__ __launch_bounds__(256)
void scores_kernel(const float* __restrict__ enc,   // [S,B,H]
                   const float* __restrict__ v,     // [B,H]
                   const float* __restrict__ c,     // [B]
                   float* __restrict__ scores) {    // [B,S]
    const int gwave = (int)((blockIdx.x * blockDim.x + threadIdx.x) >> 5);
    const int lane  = threadIdx.x & 31;
    const int s     = gwave / BATCH;
    const int b     = gwave % BATCH;

    const float* __restrict__ erow = enc + ((size_t)s * BATCH + b) * HDIM;
    const float* __restrict__ vrow = v + (size_t)b * HDIM;

    float acc = 0.f;
#pragma unroll
    for (int i = 0; i < 4; ++i) {
        const int idx = (i * 32 + lane) * 4;           // 128 float4 per row
        const v4f e  = __builtin_nontemporal_load((const v4f*)(erow + idx));
        const v4f vv = *(const v4f*)(vrow + idx);
        acc += e.x * vv.x + e.y * vv.y + e.z * vv.z + e.w * vv.w;
    }
#pragma unroll
    for (int off = 16; off > 0; off >>= 1)
        acc += __shfl_xor(acc, off, 32);

    if (lane == 0)
        scores[(size_t)b * S_LEN + s] = acc + c[b];
}

// ---------------------------------------------------------------------------
// Kernel 4: out[b,0,s] = softmax_s(scores[b,s]). One block per b.
// ---------------------------------------------------------------------------
__global__ __launch_bounds__(512)
void softmax_kernel(const float* __restrict__ scores,  // [B,S]
                    float* __restrict__ out) {          // [B,1,S]
    __shared__ float red[16];
    const int b    = blockIdx.x;
    const int t    = threadIdx.x;
    const int wave = t >> 5;
    const int lane = t & 31;
    const float* __restrict__ row = scores + (size_t)b * S_LEN;

    // --- max over S ---
    float m = -INFINITY;
    for (int i = t; i < S_LEN; i += 512) m = fmaxf(m, row[i]);
#pragma unroll
    for (int off = 16; off > 0; off >>= 1)
        m = fmaxf(m, __shfl_xor(m, off, 32));
    if (lane == 0) red[wave] = m;
    __syncthreads();
    float bm = red[0];
#pragma unroll
    for (int i = 1; i < 16; ++i) bm = fmaxf(bm, red[i]);
    __syncthreads();   // everyone done reading red[] before reuse

    // --- sum of exp ---
    float ps = 0.f;
    for (int i = t; i < S_LEN; i += 512) ps += __expf(row[i] - bm);
#pragma unroll
    for (int off = 16; off > 0; off >>= 1)
        ps += __shfl_xor(ps, off, 32);
    if (lane == 0) red[wave] = ps;
    __syncthreads();
    float bs = red[0];
#pragma unroll
    for (int i = 1; i < 16; ++i) bs += red[i];
    const float inv = 1.0f / bs;

    // --- write normalized ---
    for (int i = t; i < S_LEN; i += 512)
        out[(size_t)b * S_LEN + i] = __expf(row[i] - bm) * inv;
}

// ---------------------------------------------------------------------------
extern "C" void kernel_launch(void* const* d_in, const int* in_sizes, int n_in,
                              void* d_out, int out_size, void* d_ws, size_t ws_size,
                              hipStream_t stream) {
    const float* hidden = (const float*)d_in[0];   // [B,H]
    const float* enc    = (const float*)d_in[1];   // [S,B,H]
    const float* attn_w = (const float*)d_in[2];   // [H,H]
    const float* attn_b = (const float*)d_in[3];   // [H]
    float*       out    = (float*)d_out;           // [B,1,S]

    // workspace layout: v[B*H] | c[B] | scores[B*S]   (~590 KB)
    float* v      = (float*)d_ws;
    float* c      = v + BATCH * HDIM;
    float* scores = c + BATCH;

    // 1) v = hidden @ W  (2 x 32 WMMA tiles, one wave each)
    dim3 gV(HDIM / 16, BATCH / 16);
    precompute_v_kernel<<<gV, 32, 0, stream>>>(hidden, attn_w, v);

    // 2) c = hidden @ bias
    precompute_c_kernel<<<1, 1024, 0, stream>>>(hidden, attn_b, c);

    // 3) scores: S*B waves, 8 waves (256 threads) per block
    const int nWaves  = S_LEN * BATCH;
    const int nBlocks = nWaves / 8;
    scores_kernel<<<nBlocks, 256, 0, stream>>>(enc, v, c, scores);

    // 4) softmax over S per batch row
    softmax_kernel<<<BATCH, 512, 0, stream>>>(scores, out);
}